// CovidModel_85152021611254
// MI455X (gfx1250) — compile-verified
//
#include <hip/hip_runtime.h>

typedef float v2f __attribute__((ext_vector_type(2)));
typedef float v8f __attribute__((ext_vector_type(8)));

#define JW   10       // transition window
#define WMUP 30       // warmup length
#define FC   365      // forecast days
#define EXT  384      // padded extended series length (10 warmup + 365 + pad)
#define SCEN 8        // scenarios per block
#define WAVES 8       // waves per block (256 threads)
#define NPAIR 23      // t tile-pairs: T0 = 16*p covers t in [0, 367]

// D = A(16x4) * B(4x16) + C, f32 WMMA, K accumulated over 9 chunks (K=36):
//   k = v'*18 + tau_rel,  tau_rel in [0,17] (17 = zero pad),
//   u_ext[v][scen][tau_abs], tau_abs = T0 + t0add + tau_rel, series index t' = tau_abs - 10
//   output row m = v*8 + i  ->  y[v][T0 + t0add + i] = sum_j u[v][t-1-j] * (rho[v]*pi[v][j])
//   (tau_rel = i + 9 - j)

__global__ __launch_bounds__(256)
void covid_scan_conv_kernel(const float* __restrict__ r_t,
                            const float* __restrict__ wa,   // warmup_asymp   (2,B,30)
                            const float* __restrict__ wm,   // warmup_mild    (2,B,30)
                            const float* __restrict__ we,   // warmup_extreme (2,B,30)
                            const float* __restrict__ eps,
                            const float* __restrict__ delta,
                            const float* __restrict__ rho_M,
                            const float* __restrict__ rho_X,
                            const float* __restrict__ rho_G,
                            const float* __restrict__ pi_M,
                            const float* __restrict__ pi_X,
                            const float* __restrict__ pi_G,
                            float* __restrict__ out,
                            int B)
{
    __shared__ float U0[2 * SCEN * EXT];   // a-series, later x(extreme)-series
    __shared__ float U1[2 * SCEN * EXT];   // m(mild)-series

    const int tid  = threadIdx.x;
    const int lane = tid & 31;
    const int wv   = tid >> 5;
    const int b0   = blockIdx.x * SCEN;

    // ---- per-lane WMMA operand geometry (constant all stages) ----
    const int mrow  = lane & 15;            // A-matrix row this lane holds
    const int vout  = mrow >> 3;            // vax status of this output row
    const int iout  = mrow & 7;             // t-offset of this output row
    const int kb    = (lane & 16) ? 2 : 0;  // K sub-slot (lanes 16-31 hold K=2,3)
    const int ncol  = lane & 15;            // B/D column this lane holds
    const int scen  = ncol & 7;             // scenario within block
    const int t0add = (ncol & 8) ? 8 : 0;   // second t-subtile in columns 8..15

    int offs[9];                            // LDS B-operand offsets (add T0 per tile)
#pragma unroll
    for (int c = 0; c < 9; ++c) {
        int k0 = 4 * c + kb;
        int vp = (k0 >= 18) ? 1 : 0;
        int tr = k0 - vp * 18;
        offs[c] = (vp * SCEN + scen) * EXT + tr + t0add;   // float2 @ even idx -> 8B aligned
    }

    // ================= Phase A: warmup fill + geometric prefix-product scan =================
    {
        const int s = wv;                         // one wave per scenario
        int b = b0 + s; if (b >= B) b = B - 1;
        const float* rt = r_t + (size_t)b * FC;
        const float eps_s = eps[0];
        const float d0 = delta[0], d1 = delta[1];
        const float kappa = d0 + eps_s * d1;
        const float invT  = 1.0f / 5.8f;

        if (lane < JW) {                          // last 10 warmup entries, oldest-first
#pragma unroll
            for (int v = 0; v < 2; ++v) {
                U0[(v * SCEN + s) * EXT + lane] = wa[((size_t)v * B + b) * WMUP + (WMUP - JW + lane)];
                U1[(v * SCEN + s) * EXT + lane] = wm[((size_t)v * B + b) * WMUP + (WMUP - JW + lane)];
            }
        }
        if (lane < EXT - (JW + FC)) {             // zero tail pad (9 entries)
#pragma unroll
            for (int v = 0; v < 2; ++v) {
                U0[(v * SCEN + s) * EXT + JW + FC + lane] = 0.0f;
                U1[(v * SCEN + s) * EXT + JW + FC + lane] = 0.0f;
            }
        }

        // q_t = kappa * q_{t-1} * rt_t^invT, q_0 = p0 * rt_0^invT ; a[v,t] = delta[v]*q_t
        float C = wa[(size_t)b * WMUP + (WMUP - 1)] +
                  eps_s * wa[((size_t)B + b) * WMUP + (WMUP - 1)];
#pragma unroll 1
        for (int ch = 0; ch < 12; ++ch) {
            int t = ch * 32 + lane;
            float e = 1.0f;
            if (t < FC) {
                // rt^(1/T) = exp2( log2(rt) * (1/T) )  -> v_log_f32 + v_exp_f32
                float f = __builtin_amdgcn_exp2f(invT * __builtin_amdgcn_logf(rt[t]));
                e = (t == 0) ? f : kappa * f;
            }
            float p = e;                           // wave-inclusive product scan
#pragma unroll
            for (int off = 1; off < 32; off <<= 1) {
                float o = __shfl_up(p, off, 32);
                if (lane >= off) p *= o;
            }
            float q = C * p;
            if (t < FC) {
                U0[(0 * SCEN + s) * EXT + JW + t] = d0 * q;
                U0[(1 * SCEN + s) * EXT + JW + t] = d1 * q;
            }
            C *= __shfl(p, 31, 32);
        }
    }
    __syncthreads();

    // ---- banded Toeplitz weight operand for one conv stage (per-lane, registers) ----
    auto make_weights = [&](const float* rho, const float* pi, v2f* Wgt) {
        float rv = rho[vout];
#pragma unroll
        for (int c = 0; c < 9; ++c) {
            float w0 = 0.0f, w1 = 0.0f;
            int k0 = 4 * c + kb;
            {
                int vp = (k0 >= 18) ? 1 : 0; int tr = k0 - vp * 18;
                int j  = iout + 9 - tr;
                if (vp == vout && j >= 0 && j < JW) w0 = rv * pi[vout * JW + j];
            }
            {
                int k1 = k0 + 1;
                int vp = (k1 >= 18) ? 1 : 0; int tr = k1 - vp * 18;
                int j  = iout + 9 - tr;
                if (vp == vout && j >= 0 && j < JW) w1 = rv * pi[vout * JW + j];
            }
            v2f w; w.x = w0; w.y = w1;
            Wgt[c] = w;
        }
    };

    // ---- one conv stage: LDS series -> LDS series, 9 chained f32 WMMAs per tile ----
    auto conv_to_lds = [&](const float* src, float* dst, const v2f* Wgt) {
        const int dbase = (((lane & 16) ? 1 : 0) * SCEN + scen) * EXT + JW + t0add;
        for (int pr = wv; pr < NPAIR; pr += WAVES) {     // uniform per wave: EXEC all ones
            const int T0 = pr * 16;
            v8f acc = {0.f, 0.f, 0.f, 0.f, 0.f, 0.f, 0.f, 0.f};
#pragma unroll
            for (int c = 0; c < 9; ++c) {
                v2f bv = *(const v2f*)(src + offs[c] + T0);
                acc = __builtin_amdgcn_wmma_f32_16x16x4_f32(
                          false, Wgt[c], false, bv, (short)0, acc, false, false);
            }
#pragma unroll
            for (int r = 0; r < 8; ++r) dst[dbase + T0 + r] = acc[r];
        }
    };

    v2f Wgt[9];

    // Stage 1: mild = conv(asymp)
    make_weights(rho_M, pi_M, Wgt);
    conv_to_lds(U0, U1, Wgt);
    __syncthreads();

    // Refill extreme-warmup into U0 head (a-history no longer needed; tail pad stays 0)
    if (lane < JW) {
        const int s = wv;
        int b = b0 + s; if (b >= B) b = B - 1;
#pragma unroll
        for (int v = 0; v < 2; ++v)
            U0[(v * SCEN + s) * EXT + lane] = we[((size_t)v * B + b) * WMUP + (WMUP - JW + lane)];
    }

    // Stage 2: extreme = conv(mild)   (writes only tau >= 10 of U0, no race with refill)
    make_weights(rho_X, pi_X, Wgt);
    conv_to_lds(U1, U0, Wgt);
    __syncthreads();

    // Stage 3: ward = conv(extreme), sum both vax halves, write gw directly
    make_weights(rho_G, pi_G, Wgt);
    {
        const int b = b0 + scen;
        const bool wr = (b < B) && (lane < 16);
        for (int pr = wv; pr < NPAIR; pr += WAVES) {
            const int T0 = pr * 16;
            v8f acc = {0.f, 0.f, 0.f, 0.f, 0.f, 0.f, 0.f, 0.f};
#pragma unroll
            for (int c = 0; c < 9; ++c) {
                v2f bv = *(const v2f*)(U0 + offs[c] + T0);
                acc = __builtin_amdgcn_wmma_f32_16x16x4_f32(
                          false, Wgt[c], false, bv, (short)0, acc, false, false);
            }
#pragma unroll
            for (int r = 0; r < 8; ++r) {
                float g = acc[r] + __shfl_xor(acc[r], 16, 32);   // v0 + v1 (lane L <-> L^16)
                int t = T0 + t0add + r;
                if (wr && t < FC) out[(size_t)b * FC + t] = g;
            }
        }
    }
}

extern "C" void kernel_launch(void* const* d_in, const int* in_sizes, int n_in,
                              void* d_out, int out_size, void* d_ws, size_t ws_size,
                              hipStream_t stream) {
    const float* r_t   = (const float*)d_in[0];
    const float* wa    = (const float*)d_in[1];
    const float* wm    = (const float*)d_in[2];
    const float* we    = (const float*)d_in[3];
    const float* eps   = (const float*)d_in[4];
    const float* delta = (const float*)d_in[5];
    const float* rho_M = (const float*)d_in[6];
    const float* rho_X = (const float*)d_in[7];
    const float* rho_G = (const float*)d_in[8];
    const float* pi_M  = (const float*)d_in[9];
    const float* pi_X  = (const float*)d_in[10];
    const float* pi_G  = (const float*)d_in[11];
    float* out = (float*)d_out;

    const int B = in_sizes[0] / FC;
    const int grid = (B + SCEN - 1) / SCEN;
    covid_scan_conv_kernel<<<grid, 256, 0, stream>>>(
        r_t, wa, wm, we, eps, delta, rho_M, rho_X, rho_G, pi_M, pi_X, pi_G, out, B);
}